// MegaCRNv2_53867479826915
// MI455X (gfx1250) — compile-verified
//
#include <hip/hip_runtime.h>
#include <math.h>

// MegaCRN forward on gfx1250 (MI455X), fp32 WMMA (V_WMMA_F32_16X16X4_F32).
// Chebyshev identity terms folded into weights; T2 computed as 2*g@(g@X)-X,
// so per agcn call we run 4 big (1024x1024)@(1024xC) WMMA GEMMs + 5 weight GEMMs.

#define B_ 16
#define T_ 12
#define N_ 1024

typedef float v2f __attribute__((ext_vector_type(2)));
typedef float v8f __attribute__((ext_vector_type(8)));

// ---------------------------------------------------------------------------
// Generic batched WMMA GEMM:  C[b] (+)= alpha * A[b] @ B[b]
// A: M x K (lda, batch stride sA), B: K x Ncols (ldb, sB), C: M x Ncols.
// M is always 1024. Wave tile 64(M) x 16(N): 4 accumulators share one B frag.
// Block = 8 waves as 2(Mx64) x 4(Nx16) -> 128 x 64 tile.
// Ncols is a multiple of 16; K a multiple of 4. Branchless K-loop: waves whose
// 16-col tile is out of range clamp their column base (loads stay legal) and
// skip the store.
// ---------------------------------------------------------------------------
__global__ __launch_bounds__(256)
void k_wmma_gemm(const float* __restrict__ A, int lda, long long sA,
                 const float* __restrict__ Bm, int ldb, long long sB,
                 float* __restrict__ C, int ldc, long long sC,
                 int Ncols, int K, float alpha, int accum)
{
    const int bz = blockIdx.z;
    A  += (long long)bz * sA;
    Bm += (long long)bz * sB;
    C  += (long long)bz * sC;

    const int lane = threadIdx.x & 31;
    const int wave = threadIdx.x >> 5;
    const int mw   = wave & 1;          // M group (0/1) -> 64 rows each
    const int nw   = wave >> 1;         // N tile (0..3) -> 16 cols each
    const int m0   = blockIdx.y * 128 + mw * 64;
    const int n0   = blockIdx.x * 64 + nw * 16;
    const int mr   = lane & 15;
    const int kh   = lane >> 4;         // K-half: 0 -> K{0,1}, 1 -> K{2,3}

    const int nvalid = n0 < Ncols;          // wave-uniform
    const int ncol   = nvalid ? n0 : 0;     // clamp for safe loads

    // A fragment rows (16x4 layout: lanes 0-15 K{0,1}, lanes 16-31 K{2,3})
    const float* a0 = A + (long long)(m0 +  0 + mr) * lda + 2 * kh;
    const float* a1 = a0 + 16LL * lda;
    const float* a2 = a0 + 32LL * lda;
    const float* a3 = a0 + 48LL * lda;
    // B fragment (4x16): lanes 0-15 rows {0,1}, lanes 16-31 rows {2,3}
    const float* bp0 = Bm + (long long)(2 * kh) * ldb + ncol + mr;
    const float* bp1 = bp0 + ldb;
    const long long bstep = 4LL * ldb;

    v8f acc0 = {0.f,0.f,0.f,0.f,0.f,0.f,0.f,0.f};
    v8f acc1 = acc0, acc2 = acc0, acc3 = acc0;

    #pragma unroll 4
    for (int k = 0; k < K; k += 4) {
        v2f b;
        b.x = *bp0;
        b.y = *bp1;
        bp0 += bstep;
        bp1 += bstep;
        v2f x0 = *(const v2f*)(a0 + k);
        v2f x1 = *(const v2f*)(a1 + k);
        v2f x2 = *(const v2f*)(a2 + k);
        v2f x3 = *(const v2f*)(a3 + k);
        acc0 = __builtin_amdgcn_wmma_f32_16x16x4_f32(false, x0, false, b, (short)0, acc0, false, false);
        acc1 = __builtin_amdgcn_wmma_f32_16x16x4_f32(false, x1, false, b, (short)0, acc1, false, false);
        acc2 = __builtin_amdgcn_wmma_f32_16x16x4_f32(false, x2, false, b, (short)0, acc2, false, false);
        acc3 = __builtin_amdgcn_wmma_f32_16x16x4_f32(false, x3, false, b, (short)0, acc3, false, false);
    }

    if (!nvalid) return;
    // C/D layout: VGPR v -> row base + v + kh*8, col = n0 + (lane&15)
    v8f accs[4] = {acc0, acc1, acc2, acc3};
    #pragma unroll
    for (int i = 0; i < 4; ++i) {
        float* cp = C + (long long)(m0 + i * 16 + kh * 8) * ldc + n0 + mr;
        #pragma unroll
        for (int v = 0; v < 8; ++v) {
            float val = alpha * accs[i][v];
            if (accum) val += cp[(long long)v * ldc];
            cp[(long long)v * ldc] = val;
        }
    }
}

// ---------------------------------------------------------------------------
// Elementwise / small kernels
// ---------------------------------------------------------------------------
__global__ void k_zero(float* __restrict__ p, long long n) {
    long long i = (long long)blockIdx.x * 256 + threadIdx.x;
    if (i < n) p[i] = 0.f;
}

__global__ void k_bias(float* __restrict__ p, const float* __restrict__ b, int out, long long n) {
    long long i = (long long)blockIdx.x * 256 + threadIdx.x;
    if (i < n) p[i] = b[(int)(i % out)];
}

__global__ void k_sigmoid(float* __restrict__ p, long long n) {
    long long i = (long long)blockIdx.x * 256 + threadIdx.x;
    if (i < n) p[i] = 1.f / (1.f + expf(-p[i]));
}

// ne = We (1024x20) @ Memory (20x64)
__global__ void k_ne(float* __restrict__ ne, const float* __restrict__ We,
                     const float* __restrict__ Mem) {
    long long i = (long long)blockIdx.x * 256 + threadIdx.x;
    if (i >= (long long)N_ * 64) return;
    int n = (int)(i / 64), d = (int)(i % 64);
    float s = 0.f;
    #pragma unroll
    for (int m = 0; m < 20; ++m) s += We[n * 20 + m] * Mem[m * 64 + d];
    ne[i] = s;
}

// g1[r,c] = relu(ne1[r].ne2[c]);  g2[c,r] = same value (relu of transpose)
__global__ void k_gram(const float* __restrict__ ne1, const float* __restrict__ ne2,
                       float* __restrict__ g1, float* __restrict__ g2) {
    long long i = (long long)blockIdx.x * 256 + threadIdx.x;
    if (i >= (long long)N_ * N_) return;
    int r = (int)(i / N_), c = (int)(i % N_);
    float s = 0.f;
    #pragma unroll 8
    for (int d = 0; d < 64; ++d) s += ne1[r * 64 + d] * ne2[c * 64 + d];
    s = fmaxf(s, 0.f);
    g1[(long long)r * N_ + c] = s;
    g2[(long long)c * N_ + r] = s;
}

__global__ __launch_bounds__(256)
void k_softmax_row(float* __restrict__ G) {
    __shared__ float sm[256];
    const int tid = threadIdx.x;
    float* p = G + (long long)blockIdx.x * N_;
    float mx = -3.4e38f;
    for (int i = tid; i < N_; i += 256) mx = fmaxf(mx, p[i]);
    sm[tid] = mx; __syncthreads();
    for (int s = 128; s > 0; s >>= 1) { if (tid < s) sm[tid] = fmaxf(sm[tid], sm[tid + s]); __syncthreads(); }
    mx = sm[0]; __syncthreads();
    float sum = 0.f;
    for (int i = tid; i < N_; i += 256) { float e = expf(p[i] - mx); p[i] = e; sum += e; }
    sm[tid] = sum; __syncthreads();
    for (int s = 128; s > 0; s >>= 1) { if (tid < s) sm[tid] += sm[tid + s]; __syncthreads(); }
    const float inv = 1.f / sm[0];
    for (int i = tid; i < N_; i += 256) p[i] *= inv;
}

// Fold cheb identity terms into weights and zero-pad rows c..cpad.
// dst holds 5 matrices of cpad x out: [Wx=W0-W2+W3-W5 | W1 | W2 | W4 | W5]
__global__ void k_fold(const float* __restrict__ W, int c, int cpad, int out,
                       float* __restrict__ dst) {
    long long i = (long long)blockIdx.x * 256 + threadIdx.x;
    long long sz = (long long)cpad * out;
    if (i >= sz) return;
    int row = (int)(i / out), col = (int)(i % out);
    float w0 = 0.f, w1 = 0.f, w2 = 0.f, w3 = 0.f, w4 = 0.f, w5 = 0.f;
    if (row < c) {
        w0 = W[((long long)(0 * c + row)) * out + col];
        w1 = W[((long long)(1 * c + row)) * out + col];
        w2 = W[((long long)(2 * c + row)) * out + col];
        w3 = W[((long long)(3 * c + row)) * out + col];
        w4 = W[((long long)(4 * c + row)) * out + col];
        w5 = W[((long long)(5 * c + row)) * out + col];
    }
    dst[i]          = w0 - w2 + w3 - w5;
    dst[sz + i]     = w1;
    dst[2 * sz + i] = w2;
    dst[3 * sz + i] = w4;
    dst[4 * sz + i] = w5;
}

// encoder layer0 input build: X[b,n,:] = [x(b,t,n), (z?)*h] padded to 80
__global__ void k_build_x_enc0(float* __restrict__ X, const float* __restrict__ xin,
                               int t, const float* __restrict__ h,
                               const float* __restrict__ z) {
    long long i = (long long)blockIdx.x * 256 + threadIdx.x;
    if (i >= (long long)B_ * N_ * 80) return;
    int ch = (int)(i % 80);
    long long bn = i / 80;
    int b = (int)(bn / N_), n = (int)(bn % N_);
    float v = 0.f;
    if (ch == 0) v = xin[((long long)(b * T_ + t)) * N_ + n];
    else if (ch <= 64) {
        int j = ch - 1;
        float hv = h[bn * 64 + j];
        if (z) hv *= z[bn * 128 + j];
        v = hv;
    }
    X[i] = v;
}

// generic build for hidden-fed layers: X = [src(cx), (z?)*h(hid)] padded to cpad
__global__ void k_build_x_hid(float* __restrict__ X, int cpad,
                              const float* __restrict__ src, int cx,
                              const float* __restrict__ h, int hid,
                              const float* __restrict__ z) {
    long long i = (long long)blockIdx.x * 256 + threadIdx.x;
    if (i >= (long long)B_ * N_ * cpad) return;
    int ch = (int)(i % cpad);
    long long bn = i / cpad;
    float v = 0.f;
    if (ch < cx) v = src[bn * cx + ch];
    else if (ch < cx + hid) {
        int j = ch - cx;
        float hv = h[bn * hid + j];
        if (z) hv *= z[bn * (2LL * hid) + j];
        v = hv;
    }
    X[i] = v;
}

// decoder layer0 input build: X = [go(prev step out, 0 if t==0), ycov, (z?)*h] pad 144
__global__ void k_build_x_dec0(float* __restrict__ X, const float* __restrict__ dout,
                               int t, const float* __restrict__ yc,
                               const float* __restrict__ h, const float* __restrict__ z) {
    long long i = (long long)blockIdx.x * 256 + threadIdx.x;
    if (i >= (long long)B_ * N_ * 144) return;
    int ch = (int)(i % 144);
    long long bn = i / 144;
    int b = (int)(bn / N_), n = (int)(bn % N_);
    float v = 0.f;
    if (ch == 0) { if (t > 0) v = dout[((long long)(b * T_ + t - 1)) * N_ + n]; }
    else if (ch == 1) v = yc[((long long)(b * T_ + t)) * N_ + n];
    else if (ch < 130) {
        int j = ch - 2;
        float hv = h[bn * 128 + j];
        if (z) hv *= z[bn * 256 + j];
        v = hv;
    }
    X[i] = v;
}

// h = r*h + (1-r)*tanh(hc);  r = ZR[..., hid+j]
__global__ void k_gru(float* __restrict__ h, const float* __restrict__ zr,
                      const float* __restrict__ hc, int hid, long long n) {
    long long i = (long long)blockIdx.x * 256 + threadIdx.x;
    if (i >= n) return;
    int j = (int)(i % hid);
    long long bn = i / hid;
    float r = zr[bn * (2LL * hid) + hid + j];
    float c = tanhf(hc[i]);
    h[i] = r * h[i] + (1.f - r) * c;
}

// query = h_enc1 @ Wq (64x64)
__global__ void k_query(const float* __restrict__ h, const float* __restrict__ Wq,
                        float* __restrict__ q) {
    long long i = (long long)blockIdx.x * 256 + threadIdx.x;
    if (i >= (long long)B_ * N_ * 64) return;
    int d = (int)(i % 64);
    long long bn = i / 64;
    float s = 0.f;
    #pragma unroll 8
    for (int e = 0; e < 64; ++e) s += h[bn * 64 + e] * Wq[e * 64 + d];
    q[i] = s;
}

// attention over Memory; hd0/hd1 = concat(h, value)
__global__ void k_attval(const float* __restrict__ q, const float* __restrict__ Mem,
                         const float* __restrict__ h,
                         float* __restrict__ hd0, float* __restrict__ hd1) {
    long long bn = (long long)blockIdx.x * 256 + threadIdx.x;
    if (bn >= (long long)B_ * N_) return;
    float l[20];
    float mx = -3.4e38f;
    for (int m = 0; m < 20; ++m) {
        float s = 0.f;
        for (int d = 0; d < 64; ++d) s += q[bn * 64 + d] * Mem[m * 64 + d];
        l[m] = s;
        mx = fmaxf(mx, s);
    }
    float sum = 0.f;
    for (int m = 0; m < 20; ++m) { l[m] = expf(l[m] - mx); sum += l[m]; }
    float inv = 1.f / sum;
    for (int m = 0; m < 20; ++m) l[m] *= inv;
    for (int e = 0; e < 64; ++e) {
        float hv = h[bn * 64 + e];
        hd0[bn * 128 + e] = hv;
        hd1[bn * 128 + e] = hv;
    }
    for (int d = 0; d < 64; ++d) {
        float s = 0.f;
        for (int m = 0; m < 20; ++m) s += l[m] * Mem[m * 64 + d];
        hd0[bn * 128 + 64 + d] = s;
        hd1[bn * 128 + 64 + d] = s;
    }
}

// out[b,t,n] = h_dec1 @ proj_w + proj_b
__global__ void k_proj(const float* __restrict__ h1, const float* __restrict__ pw,
                       const float* __restrict__ pb, float* __restrict__ out, int t) {
    long long bn = (long long)blockIdx.x * 256 + threadIdx.x;
    if (bn >= (long long)B_ * N_) return;
    int b = (int)(bn / N_), n = (int)(bn % N_);
    float s = 0.f;
    #pragma unroll 8
    for (int j = 0; j < 128; ++j) s += h1[bn * 128 + j] * pw[j];
    out[((long long)(b * T_ + t)) * N_ + n] = s + pb[0];
}

// ---------------------------------------------------------------------------
extern "C" void kernel_launch(void* const* d_in, const int* in_sizes, int n_in,
                              void* d_out, int out_size, void* d_ws, size_t ws_size,
                              hipStream_t stream) {
    (void)in_sizes; (void)n_in; (void)out_size; (void)ws_size;
    const float* x       = (const float*)d_in[0];
    const float* y_cov   = (const float*)d_in[1];
    const float* Memory  = (const float*)d_in[2];
    const float* Wq      = (const float*)d_in[3];
    const float* We1     = (const float*)d_in[4];
    const float* We2     = (const float*)d_in[5];
    const float* enc0_gw = (const float*)d_in[6];
    const float* enc0_gb = (const float*)d_in[7];
    const float* enc0_uw = (const float*)d_in[8];
    const float* enc0_ub = (const float*)d_in[9];
    const float* enc1_gw = (const float*)d_in[10];
    const float* enc1_gb = (const float*)d_in[11];
    const float* enc1_uw = (const float*)d_in[12];
    const float* enc1_ub = (const float*)d_in[13];
    const float* dec0_gw = (const float*)d_in[14];
    const float* dec0_gb = (const float*)d_in[15];
    const float* dec0_uw = (const float*)d_in[16];
    const float* dec0_ub = (const float*)d_in[17];
    const float* dec1_gw = (const float*)d_in[18];
    const float* dec1_gb = (const float*)d_in[19];
    const float* dec1_uw = (const float*)d_in[20];
    const float* dec1_ub = (const float*)d_in[21];
    const float* proj_w  = (const float*)d_in[22];
    const float* proj_b  = (const float*)d_in[23];
    float* out = (float*)d_out;
    float* ws  = (float*)d_ws;

    // ---- workspace layout (floats) ----
    float* g1  = ws;
    float* g2  = g1 + (size_t)N_ * N_;
    float* ne1 = g2 + (size_t)N_ * N_;
    float* ne2 = ne1 + (size_t)N_ * 64;
    float* X   = ne2 + (size_t)N_ * 64;
    float* T1v = X   + (size_t)B_ * N_ * 256;
    float* T2v = T1v + (size_t)B_ * N_ * 256;
    float* ZR  = T2v + (size_t)B_ * N_ * 256;
    float* HC  = ZR  + (size_t)B_ * N_ * 256;
    float* he0 = HC  + (size_t)B_ * N_ * 128;
    float* he1 = he0 + (size_t)B_ * N_ * 64;
    float* hd0 = he1 + (size_t)B_ * N_ * 64;
    float* hd1 = hd0 + (size_t)B_ * N_ * 128;
    float* Wp  = hd1 + (size_t)B_ * N_ * 128;
    float* We0g = Wp;
    float* We0u = We0g + (size_t)5 *  80 * 128;
    float* We1g = We0u + (size_t)5 *  80 *  64;
    float* We1u = We1g + (size_t)5 * 128 * 128;
    float* Wd0g = We1u + (size_t)5 * 128 *  64;
    float* Wd0u = Wd0g + (size_t)5 * 144 * 256;
    float* Wd1g = Wd0u + (size_t)5 * 144 * 128;
    float* Wd1u = Wd1g + (size_t)5 * 256 * 256;

    auto NB = [](long long n) { return (unsigned)((n + 255) / 256); };

    auto GEMM = [&](const float* A, int lda, long long sA,
                    const float* Bb, int ldb, long long sB,
                    float* C, int ldc, long long sC,
                    int Ncols, int K, float alpha, int accum) {
        dim3 grid((Ncols + 63) / 64, N_ / 128, B_);
        k_wmma_gemm<<<grid, 256, 0, stream>>>(A, lda, sA, Bb, ldb, sB,
                                              C, ldc, sC, Ncols, K, alpha, accum);
    };

    // One agcn accumulation: ACC (+bias preloaded) += X@Wx + P1@W1 + 2P2@W2 + Q1@W4 + 2Q2@W5
    auto CELLG = [&](const float* Xb, int cpad, const float* Wset, int outc, float* ACC) {
        long long sX = (long long)N_ * cpad;
        long long sA = (long long)N_ * outc;
        size_t wsz = (size_t)cpad * outc;
        const float* Wx = Wset;
        const float* W1 = Wset + wsz;
        const float* W2 = Wset + 2 * wsz;
        const float* W4 = Wset + 3 * wsz;
        const float* W5 = Wset + 4 * wsz;
        GEMM(Xb, cpad, sX, Wx, outc, 0, ACC, outc, sA, outc, cpad, 1.f, 1);
        GEMM(g1, N_, 0, Xb, cpad, sX, T1v, cpad, sX, cpad, N_, 1.f, 0);
        GEMM(T1v, cpad, sX, W1, outc, 0, ACC, outc, sA, outc, cpad, 1.f, 1);
        GEMM(g1, N_, 0, T1v, cpad, sX, T2v, cpad, sX, cpad, N_, 1.f, 0);
        GEMM(T2v, cpad, sX, W2, outc, 0, ACC, outc, sA, outc, cpad, 2.f, 1);
        GEMM(g2, N_, 0, Xb, cpad, sX, T1v, cpad, sX, cpad, N_, 1.f, 0);
        GEMM(T1v, cpad, sX, W4, outc, 0, ACC, outc, sA, outc, cpad, 1.f, 1);
        GEMM(g2, N_, 0, T1v, cpad, sX, T2v, cpad, sX, cpad, N_, 1.f, 0);
        GEMM(T2v, cpad, sX, W5, outc, 0, ACC, outc, sA, outc, cpad, 2.f, 1);
    };

    const long long BN = (long long)B_ * N_;

    // ---- graph structure + weight prep ----
    k_ne<<<NB(N_ * 64), 256, 0, stream>>>(ne1, We1, Memory);
    k_ne<<<NB(N_ * 64), 256, 0, stream>>>(ne2, We2, Memory);
    k_gram<<<NB((long long)N_ * N_), 256, 0, stream>>>(ne1, ne2, g1, g2);
    k_softmax_row<<<N_, 256, 0, stream>>>(g1);
    k_softmax_row<<<N_, 256, 0, stream>>>(g2);
    k_fold<<<NB(80LL * 128), 256, 0, stream>>>(enc0_gw,  65,  80, 128, We0g);
    k_fold<<<NB(80LL * 64),  256, 0, stream>>>(enc0_uw,  65,  80,  64, We0u);
    k_fold<<<NB(128LL * 128),256, 0, stream>>>(enc1_gw, 128, 128, 128, We1g);
    k_fold<<<NB(128LL * 64), 256, 0, stream>>>(enc1_uw, 128, 128,  64, We1u);
    k_fold<<<NB(144LL * 256),256, 0, stream>>>(dec0_gw, 130, 144, 256, Wd0g);
    k_fold<<<NB(144LL * 128),256, 0, stream>>>(dec0_uw, 130, 144, 128, Wd0u);
    k_fold<<<NB(256LL * 256),256, 0, stream>>>(dec1_gw, 256, 256, 256, Wd1g);
    k_fold<<<NB(256LL * 128),256, 0, stream>>>(dec1_uw, 256, 256, 128, Wd1u);
    k_zero<<<NB(BN * 64), 256, 0, stream>>>(he0, BN * 64);
    k_zero<<<NB(BN * 64), 256, 0, stream>>>(he1, BN * 64);

    // ---- encoder: 2 stacked AGCRN layers, interleaved per step ----
    for (int t = 0; t < T_; ++t) {
        // layer 0 (cx=1, hid=64, cpad=80)
        k_build_x_enc0<<<NB(BN * 80), 256, 0, stream>>>(X, x, t, he0, nullptr);
        k_bias<<<NB(BN * 128), 256, 0, stream>>>(ZR, enc0_gb, 128, BN * 128);
        CELLG(X, 80, We0g, 128, ZR);
        k_sigmoid<<<NB(BN * 128), 256, 0, stream>>>(ZR, BN * 128);
        k_build_x_enc0<<<NB(BN * 80), 256, 0, stream>>>(X, x, t, he0, ZR);
        k_bias<<<NB(BN * 64), 256, 0, stream>>>(HC, enc0_ub, 64, BN * 64);
        CELLG(X, 80, We0u, 64, HC);
        k_gru<<<NB(BN * 64), 256, 0, stream>>>(he0, ZR, HC, 64, BN * 64);
        // layer 1 (cx=64, hid=64, cpad=128)
        k_build_x_hid<<<NB(BN * 128), 256, 0, stream>>>(X, 128, he0, 64, he1, 64, nullptr);
        k_bias<<<NB(BN * 128), 256, 0, stream>>>(ZR, enc1_gb, 128, BN * 128);
        CELLG(X, 128, We1g, 128, ZR);
        k_sigmoid<<<NB(BN * 128), 256, 0, stream>>>(ZR, BN * 128);
        k_build_x_hid<<<NB(BN * 128), 256, 0, stream>>>(X, 128, he0, 64, he1, 64, ZR);
        k_bias<<<NB(BN * 64), 256, 0, stream>>>(HC, enc1_ub, 64, BN * 64);
        CELLG(X, 128, We1u, 64, HC);
        k_gru<<<NB(BN * 64), 256, 0, stream>>>(he1, ZR, HC, 64, BN * 64);
    }

    // ---- memory attention: h_t = concat(h_enc1, att@Memory); init decoder states
    k_query<<<NB(BN * 64), 256, 0, stream>>>(he1, Wq, T1v);  // reuse T1v as query buf
    k_attval<<<NB(BN), 256, 0, stream>>>(T1v, Memory, he1, hd0, hd1);

    // ---- decoder: 2 stacked cells (hid=128) + projection per step ----
    for (int t = 0; t < T_; ++t) {
        // layer 0 (cx=2, hid=128, cpad=144)
        k_build_x_dec0<<<NB(BN * 144), 256, 0, stream>>>(X, out, t, y_cov, hd0, nullptr);
        k_bias<<<NB(BN * 256), 256, 0, stream>>>(ZR, dec0_gb, 256, BN * 256);
        CELLG(X, 144, Wd0g, 256, ZR);
        k_sigmoid<<<NB(BN * 256), 256, 0, stream>>>(ZR, BN * 256);
        k_build_x_dec0<<<NB(BN * 144), 256, 0, stream>>>(X, out, t, y_cov, hd0, ZR);
        k_bias<<<NB(BN * 128), 256, 0, stream>>>(HC, dec0_ub, 128, BN * 128);
        CELLG(X, 144, Wd0u, 128, HC);
        k_gru<<<NB(BN * 128), 256, 0, stream>>>(hd0, ZR, HC, 128, BN * 128);
        // layer 1 (cx=128, hid=128, cpad=256)
        k_build_x_hid<<<NB(BN * 256), 256, 0, stream>>>(X, 256, hd0, 128, hd1, 128, nullptr);
        k_bias<<<NB(BN * 256), 256, 0, stream>>>(ZR, dec1_gb, 256, BN * 256);
        CELLG(X, 256, Wd1g, 256, ZR);
        k_sigmoid<<<NB(BN * 256), 256, 0, stream>>>(ZR, BN * 256);
        k_build_x_hid<<<NB(BN * 256), 256, 0, stream>>>(X, 256, hd0, 128, hd1, 128, ZR);
        k_bias<<<NB(BN * 128), 256, 0, stream>>>(HC, dec1_ub, 128, BN * 128);
        CELLG(X, 256, Wd1u, 128, HC);
        k_gru<<<NB(BN * 128), 256, 0, stream>>>(hd1, ZR, HC, 128, BN * 128);
        // projection -> out[:, t]; also feeds next step's go input
        k_proj<<<NB(BN), 256, 0, stream>>>(hd1, proj_w, proj_b, out, t);
    }
}